// Generator_v1_26405458936016
// MI455X (gfx1250) — compile-verified
//
#include <hip/hip_runtime.h>
#include <hip/hip_bf16.h>

// ---------------------------------------------------------------------------
// Two-layer GCN for MI455X (gfx1250, wave32).
//   layer: D^-1/2 (A+I) D^-1/2 (X W) + b, ReLU then sigmoid.
// GEMMs use V_WMMA_F32_16X16X4_F32 (exact fp32, matches reference precision;
// GEMM is ~10 GFLOP total and never the bottleneck). Edge aggregation is
// L2-resident on MI455X (h1 = 51 MB << 192 MB L2), done with coalesced
// per-(edge,feature) float atomics (global_atomic_add_f32, no-return).
// ---------------------------------------------------------------------------

#define NN 50000
#define NE 800000
#define KD 256      // IN_DIM == HID_DIM == 256 (K for both GEMMs)
#define HD 256
#define OD 128

typedef float v2f __attribute__((ext_vector_type(2)));
typedef float v8f __attribute__((ext_vector_type(8)));

// ---- degree / normalization --------------------------------------------------
__global__ void k_deg_init(float* __restrict__ deg) {
  int n = blockIdx.x * blockDim.x + threadIdx.x;
  if (n < NN) deg[n] = 1.0f;            // self-loop contributes 1 to every node
}

__global__ void k_deg_edges(const int* __restrict__ dst, float* __restrict__ deg) {
  int e = blockIdx.x * blockDim.x + threadIdx.x;
  if (e < NE) {
    unsigned d = (unsigned)dst[e];
    if (d < NN) atomicAdd(&deg[d], 1.0f);
  }
}

__global__ void k_dinv(float* __restrict__ deg) {
  int n = blockIdx.x * blockDim.x + threadIdx.x;
  if (n < NN) {
    float d = deg[n];
    deg[n] = (d > 0.0f) ? rsqrtf(d) : 0.0f;   // in-place deg -> dinv
  }
}

__global__ void k_norm(const int* __restrict__ src, const int* __restrict__ dst,
                       const float* __restrict__ dinv, float* __restrict__ nrm) {
  int e = blockIdx.x * blockDim.x + threadIdx.x;
  if (e < NE) {
    unsigned s = (unsigned)src[e], d = (unsigned)dst[e];
    float vs = (s < NN) ? dinv[s] : 0.0f;
    float vd = (d < NN) ? dinv[d] : 0.0f;
    nrm[e] = vs * vd;
  }
}

// ---- GEMM: H[NN x NC] = X[NN x 256] * W[256 x NC] via f32 WMMA --------------
// One wave computes a 16(M) x 64(N) tile: A-fragment reused across 4
// accumulators. f32 A layout (16x4): lanes 0-15 -> K=0,1 ; lanes 16-31 -> K=2,3.
// C/D layout: VGPR i, lanes 0-15 = row i, lanes 16-31 = row i+8.
template <int NC>
__global__ void k_gemm_wmma(const float* __restrict__ X, const float* __restrict__ W,
                            float* __restrict__ H) {
  const int wave = (int)((blockIdx.x * blockDim.x + threadIdx.x) >> 5);
  const int lane = threadIdx.x & 31;
  const int tilesN = NC / 64;
  const int mt = wave / tilesN;
  const int nt = wave % tilesN;
  if (mt >= NN / 16) return;            // wave-uniform: EXEC stays all-ones

  const int m0 = mt * 16;
  const int n0 = nt * 64;
  const int half = lane >> 4;           // 0: K pair {0,1}; 1: K pair {2,3}
  const int lm   = lane & 15;

  const float* __restrict__ xrow = X + (size_t)(m0 + lm) * KD;

  v8f acc0 = {}, acc1 = {}, acc2 = {}, acc3 = {};

  for (int k0 = 0; k0 < KD; k0 += 4) {
    v2f a;
    a.x = xrow[k0 + 2 * half];
    a.y = xrow[k0 + 2 * half + 1];

    const float* __restrict__ wp = W + (size_t)(k0 + 2 * half) * NC + n0 + lm;
    v2f b0; b0.x = wp[0];      b0.y = wp[NC];
    v2f b1; b1.x = wp[16];     b1.y = wp[NC + 16];
    v2f b2; b2.x = wp[32];     b2.y = wp[NC + 32];
    v2f b3; b3.x = wp[48];     b3.y = wp[NC + 48];

    acc0 = __builtin_amdgcn_wmma_f32_16x16x4_f32(false, a, false, b0, (short)0, acc0, false, false);
    acc1 = __builtin_amdgcn_wmma_f32_16x16x4_f32(false, a, false, b1, (short)0, acc1, false, false);
    acc2 = __builtin_amdgcn_wmma_f32_16x16x4_f32(false, a, false, b2, (short)0, acc2, false, false);
    acc3 = __builtin_amdgcn_wmma_f32_16x16x4_f32(false, a, false, b3, (short)0, acc3, false, false);
  }

  #pragma unroll
  for (int i = 0; i < 8; ++i) {
    const size_t row = (size_t)(m0 + i + 8 * half);
    float* __restrict__ hp = H + row * NC + n0 + lm;
    hp[0]  = acc0[i];
    hp[16] = acc1[i];
    hp[32] = acc2[i];
    hp[48] = acc3[i];
  }
}

// ---- self-loop init: out[n][f] = h[n][f] * dinv[n]^2 (float4 vectorized) ----
template <int F>
__global__ void k_selfloop(const float* __restrict__ H, const float* __restrict__ dinv,
                           float* __restrict__ out) {
  size_t i = (size_t)blockIdx.x * blockDim.x + threadIdx.x;   // float4 index
  const size_t total = (size_t)NN * F / 4;
  if (i < total) {
    size_t node = (i * 4) / F;
    float s = dinv[node]; s = s * s;
    float4 v = ((const float4*)H)[i];
    v.x *= s; v.y *= s; v.z *= s; v.w *= s;
    ((float4*)out)[i] = v;
  }
}

// ---- edge aggregation: out[dst][f] += h[src][f] * norm[e] -------------------
// One thread per (edge, feature): a wave covers 32 consecutive features of one
// edge -> fully coalesced 128B gather + coalesced atomics. L2-resident.
template <int F, int SHIFT>
__global__ void k_edge_agg(const int* __restrict__ src, const int* __restrict__ dst,
                           const float* __restrict__ nrm, const float* __restrict__ H,
                           float* __restrict__ out) {
  const size_t total = (size_t)NE << SHIFT;
  const size_t stride = (size_t)gridDim.x * blockDim.x;
  for (size_t i = (size_t)blockIdx.x * blockDim.x + threadIdx.x; i < total; i += stride) {
    const size_t e = i >> SHIFT;
    const int f = (int)(i & (F - 1));
    const unsigned s = (unsigned)src[e];
    const unsigned d = (unsigned)dst[e];
    if (s < NN && d < NN) {
      atomicAdd(&out[(size_t)d * F + f], H[(size_t)s * F + f] * nrm[e]);
    }
  }
}

// ---- epilogues --------------------------------------------------------------
__global__ void k_bias_relu(float* __restrict__ a, const float* __restrict__ b) {
  size_t i = (size_t)blockIdx.x * blockDim.x + threadIdx.x;
  if (i < (size_t)NN * HD) {
    float v = a[i] + b[i & (HD - 1)];
    a[i] = v > 0.0f ? v : 0.0f;
  }
}

__global__ void k_bias_sigmoid(float* __restrict__ a, const float* __restrict__ b) {
  size_t i = (size_t)blockIdx.x * blockDim.x + threadIdx.x;
  if (i < (size_t)NN * OD) {
    float v = a[i] + b[i & (OD - 1)];
    a[i] = 1.0f / (1.0f + __expf(-v));
  }
}

// ---------------------------------------------------------------------------
extern "C" void kernel_launch(void* const* d_in, const int* in_sizes, int n_in,
                              void* d_out, int out_size, void* d_ws, size_t ws_size,
                              hipStream_t stream) {
  const float* x  = (const float*)d_in[0];
  const int*   ei = (const int*)d_in[1];    // [2, NE], int32 per harness mapping
  const float* W1 = (const float*)d_in[2];
  const float* b1 = (const float*)d_in[3];
  const float* W2 = (const float*)d_in[4];
  const float* b2 = (const float*)d_in[5];
  float* out = (float*)d_out;

  const int* src = ei;
  const int* dst = ei + NE;

  // workspace layout (floats): dinv[NN] | norm[NE] | h1[NN*256] | a1[NN*256]
  // (~106 MB). h1 buffer is reused for the layer-2 GEMM output (NN*128).
  float* ws   = (float*)d_ws;
  float* dinv = ws;
  float* nrm  = dinv + NN;
  float* h1   = nrm + NE;
  float* a1   = h1 + (size_t)NN * HD;

  const int T = 256;

  // normalization coefficients
  k_deg_init <<<(NN + T - 1) / T, T, 0, stream>>>(dinv);
  k_deg_edges<<<(NE + T - 1) / T, T, 0, stream>>>(dst, dinv);
  k_dinv     <<<(NN + T - 1) / T, T, 0, stream>>>(dinv);
  k_norm     <<<(NE + T - 1) / T, T, 0, stream>>>(src, dst, dinv, nrm);

  // ---- layer 1: h1 = x @ W1 ; a1 = scatter(h1) ; relu(a1 + b1) ----
  {
    const int waves = (NN / 16) * (HD / 64);          // 12500
    k_gemm_wmma<HD><<<(waves + 7) / 8, T, 0, stream>>>(x, W1, h1);
    k_selfloop<HD><<<(int)(((size_t)NN * HD / 4 + T - 1) / T), T, 0, stream>>>(h1, dinv, a1);
    k_edge_agg<HD, 8><<<32768, T, 0, stream>>>(src, dst, nrm, h1, a1);
    k_bias_relu<<<(int)(((size_t)NN * HD + T - 1) / T), T, 0, stream>>>(a1, b1);
  }

  // ---- layer 2: h2 = a1 @ W2 (into h1 buffer) ; out = scatter(h2) ; sigmoid ----
  {
    const int waves = (NN / 16) * (OD / 64);          // 6250
    k_gemm_wmma<OD><<<(waves + 7) / 8, T, 0, stream>>>(a1, W2, h1);
    k_selfloop<OD><<<(int)(((size_t)NN * OD / 4 + T - 1) / T), T, 0, stream>>>(h1, dinv, out);
    k_edge_agg<OD, 7><<<32768, T, 0, stream>>>(src, dst, nrm, h1, out);
    k_bias_sigmoid<<<(int)(((size_t)NN * OD + T - 1) / T), T, 0, stream>>>(out, b2);
  }
}